// DGLSAGEModel_18073222381928
// MI455X (gfx1250) — compile-verified
//
#include <hip/hip_runtime.h>

typedef float v2f __attribute__((ext_vector_type(2)));
typedef float v8f __attribute__((ext_vector_type(8)));

#define KC  32   // K-chunk staged in LDS
#define KCP 36   // padded row stride (dwords): 36*l mod 64 distinct for l=0..15

// ---------------------------------------------------------------------------
// Zero-fill (grid-stride) -- scratch accumulators must be zeroed every call.
// ---------------------------------------------------------------------------
__global__ void sage_zero_f32(float* __restrict__ p, long n) {
  long i = (long)blockIdx.x * blockDim.x + threadIdx.x;
  long stride = (long)gridDim.x * blockDim.x;
  for (; i < n; i += stride) p[i] = 0.0f;
}

// ---------------------------------------------------------------------------
// Edge scatter: one 64-thread block per edge (src/dst become scalar loads),
// each lane moves one float4 of the 256-float feature row and accumulates
// with hardware global_atomic_add_f32. Lane 0 bumps the degree counter.
// Prefetch the next edge's source row (global_prefetch_b8).
// ---------------------------------------------------------------------------
__global__ void sage_scatter(const float* __restrict__ H,
                             const int* __restrict__ src,
                             const int* __restrict__ dst,
                             float* __restrict__ S,
                             float* __restrict__ Deg,
                             int E, int d) {
  const int e = blockIdx.x;
  if (e >= E) return;
  const int sidx = src[e];
  const int didx = dst[e];

  if (e + 1 < E) {
    const int snext = src[e + 1];
    __builtin_prefetch(H + (size_t)snext * d + threadIdx.x * 4, 0, 0);
  }

  const float4 v = *(const float4*)(H + (size_t)sidx * d + threadIdx.x * 4);
  float* o = S + (size_t)didx * d + threadIdx.x * 4;
  __hip_atomic_fetch_add(o + 0, v.x, __ATOMIC_RELAXED, __HIP_MEMORY_SCOPE_AGENT);
  __hip_atomic_fetch_add(o + 1, v.y, __ATOMIC_RELAXED, __HIP_MEMORY_SCOPE_AGENT);
  __hip_atomic_fetch_add(o + 2, v.z, __ATOMIC_RELAXED, __HIP_MEMORY_SCOPE_AGENT);
  __hip_atomic_fetch_add(o + 3, v.w, __ATOMIC_RELAXED, __HIP_MEMORY_SCOPE_AGENT);
  if (threadIdx.x == 0) {
    __hip_atomic_fetch_add(Deg + didx, 1.0f, __ATOMIC_RELAXED, __HIP_MEMORY_SCOPE_AGENT);
  }
}

// ---------------------------------------------------------------------------
// mean = s / max(deg, 1)   (float4 per thread)
// ---------------------------------------------------------------------------
__global__ void sage_normalize(float* __restrict__ S, const float* __restrict__ Deg,
                               int n, int d) {
  const int d4 = d >> 2;
  long i = (long)blockIdx.x * blockDim.x + threadIdx.x;
  long total = (long)n * d4;
  if (i >= total) return;
  const int row = (int)(i / d4);
  const float inv = 1.0f / fmaxf(Deg[row], 1.0f);
  float4* p = (float4*)S + i;
  float4 v = *p;
  v.x *= inv; v.y *= inv; v.z *= inv; v.w *= inv;
  *p = v;
}

// ---------------------------------------------------------------------------
// Fused dual-GEMM with V_WMMA_F32_16X16X4_F32 and LDS-staged A panels.
//   Out = act( Hself @ Wself + Mean @ Wneigh + bias )
// Shapes are template parameters so the inner loop is branch-free:
//  - NTW column tiles per wave is a static unrolled loop (no exec shuffling)
//  - DOUT-constant B addressing folds to immediate global_load offsets
//  - RELU selects the epilogue statically.
// One block = one 16-row M-tile x all DOUT columns. The 16xKC chunk of BOTH
// A matrices is staged in double-buffered LDS (8x less A traffic), KCP
// padding makes the strided ds_load_b64 reads bank-conflict free.
// A layout (16x4 f32): lanes 0-15 hold K0/K1 in v[0]/v[1], lanes 16-31 K2/K3.
// B mirrored with N across lanes. C/D: VGPR r = row (half*8 + r).
// Row-clamped loads + guarded stores keep EXEC all-ones for every WMMA.
// ---------------------------------------------------------------------------
template <int D, int DOUT, int NTW, bool RELU>
__global__ void sage_gemm_wmma(const float* __restrict__ Hself,
                               const float* __restrict__ Mean,
                               const float* __restrict__ Wself,
                               const float* __restrict__ Wneigh,
                               const float* __restrict__ bias,
                               const float* __restrict__ dmask,
                               float* __restrict__ Out,
                               int n) {
  __shared__ float As[2][2][16 * KCP];   // [buffer][matrix][row*KCP + k]

  const int tid  = threadIdx.x;
  const int lane = tid & 31;
  const int wave = tid >> 5;
  const int tm   = blockIdx.y;
  const int half = lane >> 4;            // 0: lanes 0-15, 1: lanes 16-31
  const int l    = lane & 15;

  // -- cooperative staging assignment: 256 threads x one float4 each --
  const int smat = tid >> 7;             // 0 -> Hself, 1 -> Mean
  const int sidx = tid & 127;
  const int srow = sidx >> 3;            // 0..15
  const int scol = (sidx & 7) * 4;       // 0,4,...,28
  const int grow = min(tm * 16 + srow, n - 1);
  const float* sbase = (smat ? Mean : Hself) + (size_t)grow * D + scol;

  // -- accumulators (bias-initialized) --
  v8f acc[NTW];
#pragma unroll
  for (int t = 0; t < NTW; ++t) {
    const float bv = bias[(wave * NTW + t) * 16 + l];
#pragma unroll
    for (int r = 0; r < 8; ++r) acc[t][r] = bv;
  }

  constexpr int nchunks = D / KC;
  float4 stage = *(const float4*)(sbase);          // chunk 0 in flight

  for (int c = 0; c < nchunks; ++c) {
    const int buf = c & 1;
    *(float4*)&As[buf][smat][srow * KCP + scol] = stage;
    if (c + 1 < nchunks) {                         // prefetch next chunk
      stage = *(const float4*)(sbase + (size_t)(c + 1) * KC);
    }
    __syncthreads();

    // per-chunk B base pointers; k0*DOUT folds to immediate offsets
    const size_t wbase = (size_t)(c * KC + half * 2) * DOUT;
    const float* pW0[NTW];
    const float* pW1[NTW];
#pragma unroll
    for (int t = 0; t < NTW; ++t) {
      const int col = (wave * NTW + t) * 16 + l;
      pW0[t] = Wself + wbase + col;
      pW1[t] = Wneigh + wbase + col;
    }

#pragma unroll
    for (int k0 = 0; k0 < KC; k0 += 4) {
      const int kk = k0 + half * 2;
      const v2f a0 = *(const v2f*)&As[buf][0][l * KCP + kk];
      const v2f a1 = *(const v2f*)&As[buf][1][l * KCP + kk];
#pragma unroll
      for (int t = 0; t < NTW; ++t) {
        v2f b0, b1;
        b0.x = pW0[t][k0 * DOUT];
        b0.y = pW0[t][k0 * DOUT + DOUT];
        acc[t] = __builtin_amdgcn_wmma_f32_16x16x4_f32(
            false, a0, false, b0, (short)0, acc[t], false, false);
        b1.x = pW1[t][k0 * DOUT];
        b1.y = pW1[t][k0 * DOUT + DOUT];
        acc[t] = __builtin_amdgcn_wmma_f32_16x16x4_f32(
            false, a1, false, b1, (short)0, acc[t], false, false);
      }
    }
  }

  // -- epilogue: bias already folded; apply ReLU*dmask and store --
#pragma unroll
  for (int t = 0; t < NTW; ++t) {
    const int col = (wave * NTW + t) * 16 + l;
#pragma unroll
    for (int r = 0; r < 8; ++r) {
      const int row = tm * 16 + half * 8 + r;
      if (row < n) {
        float v = acc[t][r];
        if (RELU) v = fmaxf(v, 0.0f) * dmask[(size_t)row * DOUT + col];
        Out[(size_t)row * DOUT + col] = v;
      }
    }
  }
}

// ---------------------------------------------------------------------------
// Launch: 3 x (zeroed scatter-mean -> WMMA dual-GEMM) pipeline on `stream`.
// ---------------------------------------------------------------------------
extern "C" void kernel_launch(void* const* d_in, const int* in_sizes, int n_in,
                              void* d_out, int out_size, void* d_ws, size_t ws_size,
                              hipStream_t stream) {
  const float* h      = (const float*)d_in[0];
  const int*  src1    = (const int*)d_in[1];
  const int*  dst1    = (const int*)d_in[2];
  const int*  src2    = (const int*)d_in[3];
  const int*  dst2    = (const int*)d_in[4];
  const int*  src3    = (const int*)d_in[5];
  const int*  dst3    = (const int*)d_in[6];
  const float* Wself1  = (const float*)d_in[10];
  const float* Wneigh1 = (const float*)d_in[11];
  const float* b1      = (const float*)d_in[12];
  const float* Wself2  = (const float*)d_in[13];
  const float* Wneigh2 = (const float*)d_in[14];
  const float* b2      = (const float*)d_in[15];
  const float* Wself3  = (const float*)d_in[16];
  const float* Wneigh3 = (const float*)d_in[17];
  const float* b3      = (const float*)d_in[18];
  const float* dmask1  = (const float*)d_in[19];
  const float* dmask2  = (const float*)d_in[20];

  const int D = 256, Dout = 128;
  const int E1 = in_sizes[1];
  const int E2 = in_sizes[3];
  const int E3 = in_sizes[5];
  const int N1 = in_sizes[19] / D;   // dmask1 is [N1, 256]
  const int N2 = in_sizes[20] / D;   // dmask2 is [N2, 256]
  const int N3 = out_size / Dout;    // output is [N3, 128]

  // Workspace layout: [s1|deg1|s2|deg2|s3|deg3] (zeroed prefix) then h1, h2.
  float* w = (float*)d_ws;
  float* s1 = w;   w += (size_t)N1 * D;
  float* deg1 = w; w += N1;
  float* s2 = w;   w += (size_t)N2 * D;
  float* deg2 = w; w += N2;
  float* s3 = w;   w += (size_t)N3 * D;
  float* deg3 = w; w += N3;
  float* h1 = w;   w += (size_t)N1 * D;
  float* h2 = w;   w += (size_t)N2 * D;

  const long zcount = (long)(N1 + N2 + N3) * (D + 1);
  sage_zero_f32<<<2048, 256, 0, stream>>>((float*)d_ws, zcount);

  // ---- Layer 1: h (N0 x 256) -> h1 (N1 x 256) ----
  sage_scatter<<<E1, 64, 0, stream>>>(h, src1, dst1, s1, deg1, E1, D);
  {
    long t4 = (long)N1 * (D / 4);
    sage_normalize<<<(unsigned)((t4 + 255) / 256), 256, 0, stream>>>(s1, deg1, N1, D);
  }
  sage_gemm_wmma<256, 256, 2, true><<<dim3(1, (N1 + 15) / 16), 256, 0, stream>>>(
      h, s1, Wself1, Wneigh1, b1, dmask1, h1, N1);

  // ---- Layer 2: h1 -> h2 (N2 x 256) ----
  sage_scatter<<<E2, 64, 0, stream>>>(h1, src2, dst2, s2, deg2, E2, D);
  {
    long t4 = (long)N2 * (D / 4);
    sage_normalize<<<(unsigned)((t4 + 255) / 256), 256, 0, stream>>>(s2, deg2, N2, D);
  }
  sage_gemm_wmma<256, 256, 2, true><<<dim3(1, (N2 + 15) / 16), 256, 0, stream>>>(
      h1, s2, Wself2, Wneigh2, b2, dmask2, h2, N2);

  // ---- Layer 3: h2 -> out (N3 x 128), no activation ----
  sage_scatter<<<E3, 64, 0, stream>>>(h2, src3, dst3, s3, deg3, E3, D);
  {
    long t4 = (long)N3 * (D / 4);
    sage_normalize<<<(unsigned)((t4 + 255) / 256), 256, 0, stream>>>(s3, deg3, N3, D);
  }
  sage_gemm_wmma<256, 128, 1, false><<<dim3(1, (N3 + 15) / 16), 256, 0, stream>>>(
      h2, s3, Wself3, Wneigh3, b3, nullptr, (float*)d_out, N3);
}